// Concordance_Loss_1692217114660
// MI455X (gfx1250) — compile-verified
//
#include <hip/hip_runtime.h>
#include <stdint.h>

// Concordance (Harrell C-index) loss, N=8192.
// O(N^2) pairwise comparison reduction. No MAC structure -> WMMA inapplicable;
// the CDNA5-specific win is async global->LDS staging (ASYNCcnt) of j-tiles.

#define BLOCK 256   // one i per thread
#define TJ    512   // j-tile resident in LDS per block

#if defined(__AMDGCN__) && __has_builtin(__builtin_amdgcn_global_load_async_to_lds_b64)
#define HAVE_ASYNC_LDS 1
#else
#define HAVE_ASYNC_LDS 0
#endif

#define AS1 __attribute__((address_space(1)))
#define AS3 __attribute__((address_space(3)))

#if HAVE_ASYNC_LDS
// Exact pointee type from the compiler diagnostic:
// 'int __attribute__((__vector_size__(2 * sizeof(int)))) *'
typedef int v2i __attribute__((vector_size(8)));
typedef AS1 v2i* gv2i_p;   // global (device) address space pointer
typedef AS3 v2i* lv2i_p;   // LDS (shared) address space pointer
#endif

__global__ __launch_bounds__(32) void cidx_zero(unsigned int* counters) {
    if (threadIdx.x < 3) counters[threadIdx.x] = 0u;
}

__global__ __launch_bounds__(BLOCK) void cidx_pairs(
    const int*   __restrict__ ev,   // event_indicator (nonzero = true)
    const float* __restrict__ tt,   // event_time (monotone proxy for exp(time))
    const float* __restrict__ rr,   // estimate
    unsigned int* __restrict__ counters)  // [conc, tied, total]
{
    __shared__ float s_t[TJ];
    __shared__ float s_r[TJ];
    __shared__ int   s_d[TJ];
    __shared__ unsigned long long s_red[BLOCK];

    const int tid = threadIdx.x;
    const int i   = blockIdx.x * BLOCK + tid;
    const int j0  = blockIdx.y * TJ;

    // ---- stage j-tile into LDS (async copy engine on gfx1250) ----------
    {
        const int e = tid * 2;                 // 2 elements = 8 bytes / thread
#if HAVE_ASYNC_LDS
        // AS1 pointer value == flat address for global memory; AS3 pointer
        // value == low 32 bits of the flat LDS address.
        __builtin_amdgcn_global_load_async_to_lds_b64(
            (gv2i_p)(uintptr_t)(tt + j0 + e),
            (lv2i_p)(uint32_t)(uintptr_t)(&s_t[e]), 0, 0);
        __builtin_amdgcn_global_load_async_to_lds_b64(
            (gv2i_p)(uintptr_t)(rr + j0 + e),
            (lv2i_p)(uint32_t)(uintptr_t)(&s_r[e]), 0, 0);
        __builtin_amdgcn_global_load_async_to_lds_b64(
            (gv2i_p)(uintptr_t)(ev + j0 + e),
            (lv2i_p)(uint32_t)(uintptr_t)(&s_d[e]), 0, 0);
#if __has_builtin(__builtin_amdgcn_s_wait_asynccnt)
        __builtin_amdgcn_s_wait_asynccnt(0);
#else
        asm volatile("s_wait_asynccnt 0" ::: "memory");
#endif
#else
        const float2 tv = *(const float2*)(tt + j0 + e);
        const float2 rv = *(const float2*)(rr + j0 + e);
        const int2   dv = *(const int2  *)(ev + j0 + e);
        *(float2*)(&s_t[e]) = tv;
        *(float2*)(&s_r[e]) = rv;
        *(int2  *)(&s_d[e]) = dv;
#endif
    }
    __syncthreads();

    // ---- per-thread pair scan (branchless) ------------------------------
    const unsigned int di = (ev[i] != 0) ? 1u : 0u;
    const float ti = tt[i];
    const float ri = rr[i];

    unsigned int c_con = 0, c_tie = 0, c_tot = 0;
#pragma unroll 8
    for (int jj = 0; jj < TJ; ++jj) {
        const float tj = s_t[jj];           // wave-uniform LDS broadcast
        const float rj = s_r[jj];
        const unsigned int dj = (s_d[jj] != 0) ? 1u : 0u;
        // comp = d_i & ((ti<tj) | ((ti==tj) & ~d_j)); i==j self-pair is
        // automatically excluded (needs d_i & ~d_i).
        const unsigned int comp =
            di & (((ti < tj) ? 1u : 0u) | (((ti == tj) ? 1u : 0u) & (dj ^ 1u)));
        const float rd = rj - ri;
        c_tot += comp;
        c_con += comp & ((rd < 0.0f) ? 1u : 0u);
        c_tie += comp & ((fabsf(rd) <= 1e-8f) ? 1u : 0u);
    }

    // ---- block reduction: pack 3 counts (<=512 each, block sum < 2^21) --
    unsigned long long packed =
        (unsigned long long)c_con |
        ((unsigned long long)c_tie << 21) |
        ((unsigned long long)c_tot << 42);
    s_red[tid] = packed;
    __syncthreads();
#pragma unroll
    for (int s = BLOCK / 2; s > 0; s >>= 1) {
        if (tid < s) s_red[tid] += s_red[tid + s];
        __syncthreads();
    }
    if (tid == 0) {
        const unsigned long long v = s_red[0];
        atomicAdd(&counters[0], (unsigned int)( v         & 0x1FFFFFull));
        atomicAdd(&counters[1], (unsigned int)((v >> 21)  & 0x1FFFFFull));
        atomicAdd(&counters[2], (unsigned int)((v >> 42)  & 0x1FFFFFull));
    }
}

__global__ __launch_bounds__(32) void cidx_final(const unsigned int* __restrict__ c,
                                                 float* __restrict__ out)
{
    if (threadIdx.x == 0) {
        const double conc = (double)c[0];
        const double tie  = (double)c[1];
        const double tot  = (double)c[2];
        const double disc = tot - conc - tie;
        const double loss = (disc + 0.5 * tie) / (disc + conc + tie + 1e-7);
        out[0] = (float)(1.0 - loss);
    }
}

extern "C" void kernel_launch(void* const* d_in, const int* in_sizes, int n_in,
                              void* d_out, int out_size, void* d_ws, size_t ws_size,
                              hipStream_t stream) {
    const int*   ev = (const int*)  d_in[0];   // event_indicator
    const float* tt = (const float*)d_in[1];   // event_time
    const float* rr = (const float*)d_in[2];   // estimate
    unsigned int* counters = (unsigned int*)d_ws;

    const int n = in_sizes[0];                 // 8192
    cidx_zero<<<1, 32, 0, stream>>>(counters);
    dim3 grid(n / BLOCK, n / TJ);              // 32 x 16 = 512 workgroups
    cidx_pairs<<<grid, BLOCK, 0, stream>>>(ev, tt, rr, counters);
    cidx_final<<<1, 32, 0, stream>>>(counters, (float*)d_out);
}